// GivensRotation_48996986913228
// MI455X (gfx1250) — compile-verified
//
#include <hip/hip_runtime.h>

typedef __attribute__((ext_vector_type(2))) float v2f;
typedef __attribute__((ext_vector_type(4))) float v4f;
typedef __attribute__((ext_vector_type(8))) float v8f;

#define B_ 4
#define C_ 4096
#define M_ 2048
#define H_ 16
#define D_ 128
#define S_ 64

// ---------------------------------------------------------------------------
// Kernel 1: pos[b,h,c] = (x[b,c,:] . W[h,:]) + pos_weight[h]*c
// One wave computes one 16(c) x 16(h) tile with V_WMMA_F32_16X16X4_F32.
//
// fp32 WMMA operand layout (ISA 7.12.2):
//   A 16x4 (M=c,K):  lanes 0-15 row M=lane, VGPR0=K0,VGPR1=K1;
//                    lanes 16-31 row M=lane-16, VGPR0=K2,VGPR1=K3
//     -> lane loads x[c0+(lane&15), k + 2*(lane>>4)] as float2
//   B 4x16 (K,N=h):  symmetric, lane loads W[(lane&15), k + 2*(lane>>4)] float2
//   C/D 16x16: VGPR v = rows M=v (lanes 0-15) / M=v+8 (lanes 16-31), N=lane&15
// ---------------------------------------------------------------------------
__global__ __launch_bounds__(32) void pos_gemm_wmma(
    const float* __restrict__ x,      // [B,C,M]
    const float* __restrict__ w,      // [H,M]
    const float* __restrict__ pw,     // [H]
    float* __restrict__ pos)          // [B,H,C]  (workspace)
{
    const int lane = threadIdx.x;          // 0..31, full wave, EXEC all ones
    const int tile = blockIdx.x;           // 0 .. B*C/16 - 1
    const int b    = tile / (C_ / 16);
    const int c0   = (tile % (C_ / 16)) * 16;

    const int half = lane >> 4;            // 0 | 1
    const int l16  = lane & 15;

    const float* xrow = x + ((size_t)(b * C_ + c0 + l16)) * M_ + 2 * half;
    const float* wrow = w + (size_t)l16 * M_ + 2 * half;

    v8f acc0 = {};
    v8f acc1 = {};

    #pragma unroll 4
    for (int k = 0; k < M_; k += 8) {
        v2f a0 = *(const v2f*)(xrow + k);
        v2f b0 = *(const v2f*)(wrow + k);
        v2f a1 = *(const v2f*)(xrow + k + 4);
        v2f b1 = *(const v2f*)(wrow + k + 4);
        // 8 args: (neg_a, A, neg_b, B, c_mod, C, reuse_a, reuse_b)
        acc0 = __builtin_amdgcn_wmma_f32_16x16x4_f32(
            false, a0, false, b0, (short)0, acc0, false, false);
        acc1 = __builtin_amdgcn_wmma_f32_16x16x4_f32(
            false, a1, false, b1, (short)0, acc1, false, false);
    }
    v8f acc = acc0 + acc1;

    const int   h   = l16;
    const float pwh = pw[h];

    #pragma unroll
    for (int v = 0; v < 8; ++v) {
        const int c = c0 + v + 8 * half;
        pos[((size_t)b * H_ + h) * C_ + c] = acc[v] + pwh * (float)c;
    }
}

// ---------------------------------------------------------------------------
// Kernel 2: rotation. One wave per (b,h,c); lane L handles d = 4L..4L+3,
// i.e. rotation pairs s = 2L and 2L+1.  b128 loads/stores keep the memory
// pipe at line rate (512B per wave per q/k row).  q/k/out are pure streaming
// (touched exactly once) -> non-temporal hint so 537 MB of traffic doesn't
// rinse the 192 MB L2; pos/theta stay regular-temporal (heavily reused).
// ---------------------------------------------------------------------------
__global__ __launch_bounds__(256) void rotate_qk(
    const float* __restrict__ q,      // [B,H,C,D]
    const float* __restrict__ k,      // [B,H,C,D]
    const float* __restrict__ pos,    // [B,H,C]
    const float* __restrict__ theta,  // [S]
    float* __restrict__ out)          // [2,B,H,C,D]
{
    const int    lane = threadIdx.x & 31;
    const size_t bhc  = (size_t)blockIdx.x * 8 + (threadIdx.x >> 5);

    const float p = pos[bhc];                       // broadcast load (all lanes same addr)
    const v2f  th = *(const v2f*)(theta + 2 * lane);

    float s0, c0, s1, c1;
    __sincosf(p * th.x, &s0, &c0);
    __sincosf(p * th.y, &s1, &c1);

    const size_t base = bhc * (size_t)D_ + 4 * lane;
    const v4f qv = __builtin_nontemporal_load((const v4f*)(q + base));
    const v4f kv = __builtin_nontemporal_load((const v4f*)(k + base));

    v4f rq, rk;
    rq.x = c0 * qv.x - s0 * qv.y;  rq.y = s0 * qv.x + c0 * qv.y;
    rq.z = c1 * qv.z - s1 * qv.w;  rq.w = s1 * qv.z + c1 * qv.w;
    rk.x = c0 * kv.x - s0 * kv.y;  rk.y = s0 * kv.x + c0 * kv.y;
    rk.z = c1 * kv.z - s1 * kv.w;  rk.w = s1 * kv.z + c1 * kv.w;

    const size_t N = (size_t)B_ * H_ * C_ * D_;
    __builtin_nontemporal_store(rq, (v4f*)(out + base));
    __builtin_nontemporal_store(rk, (v4f*)(out + N + base));
}

extern "C" void kernel_launch(void* const* d_in, const int* in_sizes, int n_in,
                              void* d_out, int out_size, void* d_ws, size_t ws_size,
                              hipStream_t stream) {
    const float* x  = (const float*)d_in[0];   // [B,C,M]
    const float* q  = (const float*)d_in[1];   // [B,H,C,D]
    const float* k  = (const float*)d_in[2];   // [B,H,C,D]
    const float* w  = (const float*)d_in[3];   // [H,M]
    const float* pw = (const float*)d_in[4];   // [H]
    const float* th = (const float*)d_in[5];   // [S]
    float* out = (float*)d_out;
    float* pos = (float*)d_ws;                 // B*H*C floats = 1 MB

    pos_gemm_wmma<<<(B_ * C_) / 16, 32, 0, stream>>>(x, w, pw, pos);
    rotate_qk<<<(B_ * H_ * C_) / 8, 256, 0, stream>>>(q, k, pos, th, out);
}